// GIN_8641474200054
// MI455X (gfx1250) — compile-verified
//
#include <hip/hip_runtime.h>
#include <cstdint>

typedef __attribute__((ext_vector_type(2))) float v2f;
typedef __attribute__((ext_vector_type(8))) float v8f;

#define D 64
#define GIN_BN_EPS 1e-5f
// Swizzled-W pitch in dwords: 128 payload + 32 pad so consecutive k-pairs sit
// 32 banks apart -> the two half-wave groups of a b64 fragment load hit
// disjoint bank sets (lanes 0-15 banks 0-31, lanes 16-31 banks 32-63).
#define WPITCH 160

// ---------------------------------------------------------------------------
// agg = x  (init), then agg[dst] += x[src] per edge via float atomics
// ---------------------------------------------------------------------------
__global__ void gin_copy_init_kernel(const float* __restrict__ x,
                                     float* __restrict__ agg, int n) {
  int i = blockIdx.x * blockDim.x + threadIdx.x;
  if (i < n) agg[i] = x[i];
}

__global__ void gin_scatter_add_kernel(const float* __restrict__ x,
                                       const long long* __restrict__ src,
                                       const long long* __restrict__ dst,
                                       float* __restrict__ agg,
                                       long long total) {
  long long i = (long long)blockIdx.x * blockDim.x + threadIdx.x;
  if (i >= total) return;              // total = E * 32 (2 features / thread)
  const int e = (int)(i >> 5);
  const int f = (int)(i & 31) * 2;
  const long long s = src[e];
  const long long d = dst[e];
  const float2 v = *(const float2*)(x + s * D + f);
  float* p = agg + d * D + f;
  atomicAdd(p, v.x);
  atomicAdd(p + 1, v.y);
}

// ---------------------------------------------------------------------------
// Z = (BN_IN ? relu(A*scale_in+shift_in) : A) @ W + bias      [N x 64]
// Accumulates per-column sum / sum-of-squares of Z for BatchNorm.
// Block = 256 threads = 8 wave32s; each wave computes two 16x64 strips with
// v_wmma_f32_16x16x4_f32. W is staged in LDS pre-swizzled so every B fragment
// is a single aligned ds_load_b64 feeding WMMA directly.
// ---------------------------------------------------------------------------
template <bool BN_IN>
__global__ __launch_bounds__(256)
void gin_gemm_bn_stats_kernel(const float* __restrict__ A,
                              const float* __restrict__ W,
                              const float* __restrict__ bias,
                              const float* __restrict__ scale_in,
                              const float* __restrict__ shift_in,
                              float* __restrict__ Z,
                              float* __restrict__ colsum,
                              float* __restrict__ colsumsq,
                              int N) {
  __shared__ float sWs[32 * WPITCH];  // swizzled: sWs[(k>>1)*WPITCH + c*2 + (k&1)]
  __shared__ float sBias[D];
  __shared__ float sScale[D];
  __shared__ float sShift[D];
  __shared__ float sSum[D];
  __shared__ float sSumSq[D];

  const int tid = threadIdx.x;

  // ---- Stage W swizzled: thread (p, c0) interleaves rows 2p and 2p+1 so
  //      the pair {W[k][c], W[k+1][c]} is contiguous (one ds_store_b64 each).
  {
    const int p = tid >> 3;          // k-pair index 0..31
    const int c0 = (tid & 7) * 8;    // column group base
    const float4* r0 = (const float4*)(W + (2 * p) * D + c0);
    const float4* r1 = (const float4*)(W + (2 * p + 1) * D + c0);
    const float4 e0 = r0[0], e1 = r0[1];
    const float4 o0 = r1[0], o1 = r1[1];
    float2* d2 = (float2*)(sWs + p * WPITCH + c0 * 2);
    d2[0] = make_float2(e0.x, o0.x);
    d2[1] = make_float2(e0.y, o0.y);
    d2[2] = make_float2(e0.z, o0.z);
    d2[3] = make_float2(e0.w, o0.w);
    d2[4] = make_float2(e1.x, o1.x);
    d2[5] = make_float2(e1.y, o1.y);
    d2[6] = make_float2(e1.z, o1.z);
    d2[7] = make_float2(e1.w, o1.w);
  }
  if (tid < D) {
    sBias[tid] = bias[tid];
    sSum[tid] = 0.f;
    sSumSq[tid] = 0.f;
    if (BN_IN) {
      sScale[tid] = scale_in[tid];
      sShift[tid] = shift_in[tid];
    }
  }
  __syncthreads();

  const int wave = tid >> 5;
  const int lane = tid & 31;
  const int m = lane & 15;    // row within tile (A) / col within tile (B,C)
  const int half = lane >> 4; // selects K pair (A,B) / row group (C)

  // Two 16-row tiles per wave: block covers 256 rows.
  for (int rt = 0; rt < 2; ++rt) {
    const int rowBase = blockIdx.x * 256 + rt * 128 + wave * 16;
    if (rowBase >= N) break;  // wave-uniform: EXEC stays all-ones inside

    // ---- Preload A fragments (rows clamped; invalid rows masked at store).
    int arow = rowBase + m;
    if (arow > N - 1) arow = N - 1;
    const float* Arow = A + (long long)arow * D + 2 * half;
    v2f areg[16];
#pragma unroll
    for (int t = 0; t < 16; ++t) {
      float2 v = *(const float2*)(Arow + 4 * t);
      if (BN_IN) {
        const int k = 4 * t + 2 * half;
        v.x = fmaxf(fmaf(v.x, sScale[k], sShift[k]), 0.f);
        v.y = fmaxf(fmaf(v.y, sScale[k + 1], sShift[k + 1]), 0.f);
      }
      areg[t].x = v.x;
      areg[t].y = v.y;
    }

    v8f c[4];
#pragma unroll
    for (int ct = 0; ct < 4; ++ct)
#pragma unroll
      for (int i = 0; i < 8; ++i) c[ct][i] = 0.f;

    // ---- 64 x v_wmma_f32_16x16x4_f32; B fragment = one aligned ds_load_b64.
#pragma unroll
    for (int t = 0; t < 16; ++t) {
      const float* brow = sWs + (2 * t + half) * WPITCH;
#pragma unroll
      for (int ct = 0; ct < 4; ++ct) {
        const v2f b = *(const v2f*)(brow + (ct * 16 + m) * 2);
        c[ct] = __builtin_amdgcn_wmma_f32_16x16x4_f32(
            false, areg[t], false, b, (short)0, c[ct], false, false);
      }
    }

    // ---- bias, store, per-column stats ----
#pragma unroll
    for (int ct = 0; ct < 4; ++ct) {
      const int col = ct * 16 + m;
      const float bv = sBias[col];
      float s = 0.f, sq = 0.f;
#pragma unroll
      for (int i = 0; i < 8; ++i) {
        const int r = rowBase + half * 8 + i;
        const float v = c[ct][i] + bv;
        if (r < N) {
          Z[(long long)r * D + col] = v;
          s += v;
          sq += v * v;
        }
      }
      atomicAdd(&sSum[col], s);
      atomicAdd(&sSumSq[col], sq);
    }
  }

  __syncthreads();
  if (tid < D) {
    atomicAdd(&colsum[tid], sSum[tid]);
    atomicAdd(&colsumsq[tid], sSumSq[tid]);
  }
}

// ---------------------------------------------------------------------------
// scale = gamma * rsqrt(var + eps); shift = beta - mean * scale
// ---------------------------------------------------------------------------
__global__ void gin_finalize_stats_kernel(const float* __restrict__ sum,
                                          const float* __restrict__ sumsq,
                                          const float* __restrict__ gamma,
                                          const float* __restrict__ beta,
                                          float* __restrict__ scale,
                                          float* __restrict__ shift,
                                          float invN) {
  const int cidx = threadIdx.x;
  if (cidx < D) {
    const float mean = sum[cidx] * invN;
    const float var = sumsq[cidx] * invN - mean * mean;
    const float sc = gamma[cidx] * rsqrtf(var + GIN_BN_EPS);
    scale[cidx] = sc;
    shift[cidx] = fmaf(-mean, sc, beta[cidx]);
  }
}

__global__ void gin_bn_apply_kernel(float* __restrict__ Z,
                                    const float* __restrict__ scale,
                                    const float* __restrict__ shift,
                                    long long total, int relu) {
  long long i = (long long)blockIdx.x * blockDim.x + threadIdx.x;
  if (i >= total) return;
  const int col = (int)(i & 63);
  float v = fmaf(Z[i], scale[col], shift[col]);
  if (relu) v = fmaxf(v, 0.f);
  Z[i] = v;
}

__global__ void gin_zero_stats_kernel(float* p) { p[threadIdx.x] = 0.f; }

// ---------------------------------------------------------------------------
// Launcher
// ---------------------------------------------------------------------------
extern "C" void kernel_launch(void* const* d_in, const int* in_sizes, int n_in,
                              void* d_out, int out_size, void* d_ws,
                              size_t ws_size, hipStream_t stream) {
  const float* x = (const float*)d_in[0];
  const long long* adj = (const long long*)d_in[1]; // int64 [2, E]
  const float* W1 = (const float*)d_in[2];
  const float* b1 = (const float*)d_in[3];
  const float* g1 = (const float*)d_in[4];
  const float* be1 = (const float*)d_in[5];
  const float* W2 = (const float*)d_in[6];
  const float* b2 = (const float*)d_in[7];
  const float* g2 = (const float*)d_in[8];
  const float* be2 = (const float*)d_in[9];
  float* out = (float*)d_out;

  const int N = in_sizes[0] / D; // 50000
  const int E = in_sizes[1] / 2; // 800000
  const long long ND = (long long)N * D;

  float* ws = (float*)d_ws;
  float* agg = ws;            // N*64
  float* Z1 = ws + ND;        // N*64
  float* stats = ws + 2 * ND; // 512 floats
  float* sum1 = stats;
  float* sumsq1 = stats + 64;
  float* sum2 = stats + 128;
  float* sumsq2 = stats + 192;
  float* scale1 = stats + 256;
  float* shift1 = stats + 320;
  float* scale2 = stats + 384;
  float* shift2 = stats + 448;

  const int elemThreads = 256;
  const unsigned copyBlocks = (unsigned)((ND + elemThreads - 1) / elemThreads);
  gin_copy_init_kernel<<<copyBlocks, elemThreads, 0, stream>>>(x, agg, (int)ND);

  const long long edgeTotal = (long long)E * 32; // 2 features per thread
  const unsigned edgeBlocks =
      (unsigned)((edgeTotal + elemThreads - 1) / elemThreads);
  gin_scatter_add_kernel<<<edgeBlocks, elemThreads, 0, stream>>>(
      x, adj, adj + E, agg, edgeTotal);

  const unsigned gemmBlocks = (unsigned)((N + 255) / 256);
  const float invN = 1.0f / (float)N;

  for (int l = 0; l < 5; ++l) {
    float* Zout = out + (long long)l * ND;

    gin_zero_stats_kernel<<<1, 256, 0, stream>>>(stats);

    // Layer GEMM 1: Z1 = agg @ W1[l] + b1[l], accumulate BN1 stats
    gin_gemm_bn_stats_kernel<false><<<gemmBlocks, 256, 0, stream>>>(
        agg, W1 + (long long)l * D * D, b1 + l * D, nullptr, nullptr, Z1, sum1,
        sumsq1, N);

    gin_finalize_stats_kernel<<<1, 64, 0, stream>>>(
        sum1, sumsq1, g1 + l * D, be1 + l * D, scale1, shift1, invN);

    // Layer GEMM 2: Zout = relu(BN1(Z1)) @ W2[l] + b2[l], accumulate BN2 stats
    gin_gemm_bn_stats_kernel<true><<<gemmBlocks, 256, 0, stream>>>(
        Z1, W2 + (long long)l * D * D, b2 + l * D, scale1, shift1, Zout, sum2,
        sumsq2, N);

    gin_finalize_stats_kernel<<<1, 64, 0, stream>>>(
        sum2, sumsq2, g2 + l * D, be2 + l * D, scale2, shift2, invN);

    // Outer BN (+ReLU for all but last layer), in place on output slice
    gin_bn_apply_kernel<<<copyBlocks, elemThreads, 0, stream>>>(
        Zout, scale2, shift2, ND, (l < 4) ? 1 : 0);
  }
}